// CELoss_40424232190041
// MI455X (gfx1250) — compile-verified
//
#include <hip/hip_runtime.h>
#include <hip/hip_bf16.h>

// Problem constants (reference: B=2048, C=32000, T=16384; B/T derived from sizes)
#define C_DIM     32000
#define K1_T      320          // 10 waves of 32
#define CHUNK     1280         // K1_T * 4 floats per chunk
#define NCHUNK    25           // C_DIM / CHUNK == 25 exactly
#define NWAVES    (K1_T / 32)
#define NEG_BIG   (-1.0e30f)
#define LOG2E     1.44269504088896340736f
#define LN2       0.69314718055994530942f

#if defined(__HIP_DEVICE_COMPILE__) && defined(__has_builtin)
#  if __has_builtin(__builtin_amdgcn_global_load_async_to_lds_b128)
#    define HAVE_ASYNC_LDS 1
#  endif
#endif

#if defined(HAVE_ASYNC_LDS)
// Builtin signature (from clang diagnostic): (v4i addrspace(1)*, v4i addrspace(3)*, Imm int, Imm int)
typedef int v4i __attribute__((vector_size(16)));
typedef __attribute__((address_space(1))) v4i* gv4i_p;
typedef __attribute__((address_space(3))) v4i* lv4i_p;

__device__ __forceinline__ void async_cp16(const float* g, float* l) {
    __builtin_amdgcn_global_load_async_to_lds_b128(
        (gv4i_p)(g), (lv4i_p)(l), /*imm offset*/0, /*cpol*/0);
}
template <int N>
__device__ __forceinline__ void wait_async_le() {
#  if __has_builtin(__builtin_amdgcn_s_wait_asynccnt)
    __builtin_amdgcn_s_wait_asynccnt(N);
#  else
    asm volatile("s_wait_asynccnt %0" :: "i"(N) : "memory");
#  endif
    asm volatile("" ::: "memory");     // keep LDS reads behind the wait
}
#endif

__device__ __forceinline__ float exp2_fast(float x) {
#if defined(__HIP_DEVICE_COMPILE__) && defined(__has_builtin)
#  if __has_builtin(__builtin_amdgcn_exp2f)
    return __builtin_amdgcn_exp2f(x);  // v_exp_f32 IS 2^x
#  else
    return exp2f(x);
#  endif
#else
    return exp2f(x);
#endif
}
__device__ __forceinline__ float log2_fast(float x) {
#if defined(__HIP_DEVICE_COMPILE__) && defined(__has_builtin)
#  if __has_builtin(__builtin_amdgcn_logf)
    return __builtin_amdgcn_logf(x);   // v_log_f32 IS log2(x)
#  else
    return log2f(x);
#  endif
#else
    return log2f(x);
#endif
}

// ---------------- Kernel 1: per-row two-phase logsumexp -> lse[B] -------------
// Whole row (128 KB) staged in LDS with async b128 loads: HBM read exactly once.
// Each thread consumes only the bytes it requested -> no barriers while streaming.
__global__ __launch_bounds__(K1_T)
void row_lse_kernel(const float* __restrict__ preds, float* __restrict__ lse) {
    const int tid = threadIdx.x;
    const long long row = blockIdx.x;
    const float* rowp = preds + row * (long long)C_DIM;

    __shared__ __align__(16) float buf[C_DIM];          // 128 KB of the 320 KB WGP LDS
    __shared__ float rm[NWAVES], rs[NWAVES];

    float m0 = NEG_BIG, m1 = NEG_BIG, m2 = NEG_BIG, m3 = NEG_BIG;

#if defined(HAVE_ASYNC_LDS)
    // Issue all 25 async chunk loads (25 outstanding per wave; ASYNCcnt is 6-bit)
    #pragma unroll
    for (int c = 0; c < NCHUNK; ++c)
        async_cp16(rowp + c * CHUNK + tid * 4, &buf[c * CHUNK + tid * 4]);

    // Pass 1 (max), pipelined against the in-flight loads via incremental waits.
    #define P1_STEP(c)                                                      \
    {                                                                       \
        wait_async_le<(NCHUNK - 1 - (c))>();                                \
        const float4 v = *(const float4*)&buf[(c) * CHUNK + tid * 4];       \
        m0 = fmaxf(m0, v.x); m1 = fmaxf(m1, v.y);                           \
        m2 = fmaxf(m2, v.z); m3 = fmaxf(m3, v.w);                           \
    }
    P1_STEP(0)  P1_STEP(1)  P1_STEP(2)  P1_STEP(3)  P1_STEP(4)
    P1_STEP(5)  P1_STEP(6)  P1_STEP(7)  P1_STEP(8)  P1_STEP(9)
    P1_STEP(10) P1_STEP(11) P1_STEP(12) P1_STEP(13) P1_STEP(14)
    P1_STEP(15) P1_STEP(16) P1_STEP(17) P1_STEP(18) P1_STEP(19)
    P1_STEP(20) P1_STEP(21) P1_STEP(22) P1_STEP(23) P1_STEP(24)
    #undef P1_STEP
#else
    // Fallback: copy through LDS with plain loads/stores
    #pragma unroll
    for (int c = 0; c < NCHUNK; ++c) {
        const float4 v = *(const float4*)(rowp + c * CHUNK + tid * 4);
        *(float4*)&buf[c * CHUNK + tid * 4] = v;
        m0 = fmaxf(m0, v.x); m1 = fmaxf(m1, v.y);
        m2 = fmaxf(m2, v.z); m3 = fmaxf(m3, v.w);
    }
#endif

    float m = fmaxf(fmaxf(m0, m1), fmaxf(m2, m3));
    #pragma unroll
    for (int off = 16; off > 0; off >>= 1)
        m = fmaxf(m, __shfl_xor(m, off, 32));
    const int wid  = tid >> 5;
    const int lane = tid & 31;
    if (lane == 0) rm[wid] = m;
    __syncthreads();
    float M = rm[0];
    #pragma unroll
    for (int w = 1; w < NWAVES; ++w) M = fmaxf(M, rm[w]);

    // Pass 2: sum of exp. exp(x-M) = 2^(x*log2e - M*log2e): one FMA + one v_exp.
    const float nML = -M * LOG2E;
    float s0 = 0.f, s1 = 0.f, s2 = 0.f, s3 = 0.f;
    #pragma unroll
    for (int c = 0; c < NCHUNK; ++c) {
        const float4 v = *(const float4*)&buf[c * CHUNK + tid * 4];
        s0 += exp2_fast(fmaf(v.x, LOG2E, nML));
        s1 += exp2_fast(fmaf(v.y, LOG2E, nML));
        s2 += exp2_fast(fmaf(v.z, LOG2E, nML));
        s3 += exp2_fast(fmaf(v.w, LOG2E, nML));
    }
    float s = (s0 + s1) + (s2 + s3);
    #pragma unroll
    for (int off = 16; off > 0; off >>= 1)
        s += __shfl_xor(s, off, 32);
    if (lane == 0) rs[wid] = s;
    __syncthreads();
    if (tid == 0) {
        float S = rs[0];
        #pragma unroll
        for (int w = 1; w < NWAVES; ++w) S += rs[w];
        lse[row] = M + LN2 * log2_fast(S);
    }
}

// ------------- Kernel 2: gather nll terms, per-block partial sums -------------
__global__ __launch_bounds__(256)
void gather_partial_kernel(const float* __restrict__ preds,
                           const int* __restrict__ tgts,
                           const int* __restrict__ rows,
                           const float* __restrict__ lse,
                           float* __restrict__ partials, int T) {
    __shared__ float sm[256];
    const int t = blockIdx.x * 256 + threadIdx.x;
    float v = 0.0f;
    if (t < T) {
        const int r = rows[t];
        const int c = tgts[t];
        v = lse[r] - preds[(long long)r * C_DIM + c];   // -log_softmax gathered
    }
    sm[threadIdx.x] = v;
    __syncthreads();
    #pragma unroll
    for (int off = 128; off > 0; off >>= 1) {
        if (threadIdx.x < off) sm[threadIdx.x] += sm[threadIdx.x + off];
        __syncthreads();
    }
    if (threadIdx.x == 0) partials[blockIdx.x] = sm[0];
}

// ---------------- Kernel 3: deterministic final reduce + scale ----------------
__global__ __launch_bounds__(256)
void finalize_kernel(const float* __restrict__ partials, int n,
                     float* __restrict__ out, float inv_b) {
    __shared__ float sm[256];
    float v = 0.0f;
    for (int i = threadIdx.x; i < n; i += 256) v += partials[i];
    sm[threadIdx.x] = v;
    __syncthreads();
    #pragma unroll
    for (int off = 128; off > 0; off >>= 1) {
        if (threadIdx.x < off) sm[threadIdx.x] += sm[threadIdx.x + off];
        __syncthreads();
    }
    if (threadIdx.x == 0) out[0] = sm[0] * inv_b;
}

extern "C" void kernel_launch(void* const* d_in, const int* in_sizes, int n_in,
                              void* d_out, int out_size, void* d_ws, size_t ws_size,
                              hipStream_t stream) {
    const float* preds = (const float*)d_in[0];
    const int*   tgts  = (const int*)d_in[1];
    const int*   rows  = (const int*)d_in[2];
    float*       out   = (float*)d_out;

    const int B = in_sizes[0] / C_DIM;     // 2048
    const int T = in_sizes[1];             // 16384

    float* lse      = (float*)d_ws;        // [B]
    float* partials = lse + B;             // [nb2]
    const int nb2 = (T + 255) / 256;       // 64

    row_lse_kernel<<<dim3(B), dim3(K1_T), 0, stream>>>(preds, lse);
    gather_partial_kernel<<<dim3(nb2), dim3(256), 0, stream>>>(preds, tgts, rows, lse, partials, T);
    finalize_kernel<<<dim3(1), dim3(256), 0, stream>>>(partials, nb2, out, 1.0f / (float)B);
}